// DynamicPartitionStitchModule_11098195493300
// MI455X (gfx1250) — compile-verified
//
#include <hip/hip_runtime.h>
#include <hip/hip_bf16.h>
#include <stdint.h>

// ---------------------------------------------------------------------------
// DynamicPartition + DynamicStitch, N=1048576 rows x D=64 f32.
// Memory-bound scatter: ~0.52 GB total traffic -> ~22us at 23.3 TB/s.
// CDNA5 path: rows streamed global->LDS->global with the gfx1250 async
// DMA ops (ASYNCcnt), 16B/lane, non-temporal, scattered stores issued as
// contiguous 256B rows (16 lanes x B128 per row).
// ---------------------------------------------------------------------------

#define N_TOTAL 1048576
#define HALF    (N_TOTAL / 2)
#define TILE    1024           // rows per block
#define NBLOCKS (N_TOTAL / TILE)
#define THREADS 256

// --- gfx1250 async LDS <-> global helpers (inline asm; portable across
// ROCm 7.2 clang-22 and amdgpu-toolchain clang-23 builtin arities) ---------
__device__ __forceinline__ void async_load_b128(const void* gsrc, unsigned lds_off) {
    // LDS address = LDS base + byte offset held in the first VGPR; vaddr = 64-bit global address
    asm volatile("global_load_async_to_lds_b128 %0, %1, off th:TH_LOAD_NT"
                 :: "v"(lds_off), "v"(gsrc) : "memory");
}
__device__ __forceinline__ void async_store_b128(void* gdst, unsigned lds_off) {
    asm volatile("global_store_async_from_lds_b128 %0, %1, off th:TH_STORE_NT"
                 :: "v"(gdst), "v"(lds_off) : "memory");
}
__device__ __forceinline__ void wait_async0() {
    asm volatile("s_wait_asynccnt 0x0" ::: "memory");
}

// LDS generic pointer -> 32-bit LDS byte offset (low 32 bits per ISA 10.2)
__device__ __forceinline__ unsigned lds_off_of(const void* p) {
    return (unsigned)(uintptr_t)p;
}

__device__ __forceinline__ int clampi(int x, int hi) {
    return x > hi ? hi : x;
}

// ---------------------------------------------------------------------------
// Pass 1: per-block zero counts
// ---------------------------------------------------------------------------
__global__ __launch_bounds__(THREADS)
void count_kernel(const int* __restrict__ parts, int* __restrict__ blockSums) {
    __shared__ int red[THREADS];
    const int tid  = threadIdx.x;
    const int base = blockIdx.x * TILE;
    int4 v = ((const int4*)(parts + base))[tid];
    int cnt = (v.x == 0) + (v.y == 0) + (v.z == 0) + (v.w == 0);
    red[tid] = cnt;
    __syncthreads();
    for (int s = THREADS / 2; s > 0; s >>= 1) {
        if (tid < s) red[tid] += red[tid + s];
        __syncthreads();
    }
    if (tid == 0) blockSums[blockIdx.x] = red[0];
}

// ---------------------------------------------------------------------------
// Pass 2: exclusive scan of 1024 block sums (single block)
// ---------------------------------------------------------------------------
__global__ __launch_bounds__(THREADS)
void scan_kernel(const int* __restrict__ blockSums, int* __restrict__ blockOffsets) {
    __shared__ int tsum[THREADS];
    const int tid = threadIdx.x;
    int4 v = ((const int4*)blockSums)[tid];
    int s = v.x + v.y + v.z + v.w;
    tsum[tid] = s;
    __syncthreads();
    for (int off = 1; off < THREADS; off <<= 1) {
        int t = (tid >= off) ? tsum[tid - off] : 0;
        __syncthreads();
        tsum[tid] += t;
        __syncthreads();
    }
    int excl = (tid == 0) ? 0 : tsum[tid - 1];
    int4 o;
    o.x = excl;
    o.y = o.x + v.x;
    o.z = o.y + v.y;
    o.w = o.z + v.z;
    ((int4*)blockOffsets)[tid] = o;
}

// ---------------------------------------------------------------------------
// Pass 3: compute destinations, scatter rows via async LDS staging
// ---------------------------------------------------------------------------
#define PIPE 8                               // async B128s in flight per thread
#define CHUNKS_PER_OUTER (THREADS * PIPE)    // 2048 float4 chunks = 128 rows
#define OUTERS (TILE * 16 / CHUNKS_PER_OUTER)

__global__ __launch_bounds__(THREADS)
void scatter_kernel(const float* __restrict__ data,
                    const int*   __restrict__ parts,
                    const int*   __restrict__ idx0,
                    const int*   __restrict__ idx1,
                    const int*   __restrict__ blockOffsets,
                    float*       __restrict__ out) {
    __shared__ int   dests[TILE];
    __shared__ int   tscan[THREADS];
    __shared__ float stage[CHUNKS_PER_OUTER * 4]; // 32 KB staging

    const int tid  = threadIdx.x;
    const int b    = blockIdx.x;
    const int base = b * TILE;

    // --- local flags + block-wide exclusive zero-count scan ---
    int4 v = ((const int4*)(parts + base))[tid];
    const int f0 = (v.x == 0), f1 = (v.y == 0), f2 = (v.z == 0), f3 = (v.w == 0);
    tscan[tid] = f0 + f1 + f2 + f3;
    __syncthreads();
    for (int off = 1; off < THREADS; off <<= 1) {
        int t = (tid >= off) ? tscan[tid - off] : 0;
        __syncthreads();
        tscan[tid] += t;
        __syncthreads();
    }
    // exclusive zeros before first element owned by this thread
    int zb = blockOffsets[b] + ((tid == 0) ? 0 : tscan[tid - 1]);
    const int i = base + tid * 4;

    // rank0 = exclusive zero count; rank1 = i - exclusive zero count.
    // Clamp like the reference (jnp.clip) — also prevents a speculative
    // 4-byte OOB gather if the compiler evaluates both ternary arms.
    dests[4 * tid + 0] = f0 ? idx0[clampi(zb, HALF - 1)]
                            : idx1[clampi((i + 0) - zb, HALF - 1)]; zb += f0;
    dests[4 * tid + 1] = f1 ? idx0[clampi(zb, HALF - 1)]
                            : idx1[clampi((i + 1) - zb, HALF - 1)]; zb += f1;
    dests[4 * tid + 2] = f2 ? idx0[clampi(zb, HALF - 1)]
                            : idx1[clampi((i + 2) - zb, HALF - 1)]; zb += f2;
    dests[4 * tid + 3] = f3 ? idx0[clampi(zb, HALF - 1)]
                            : idx1[clampi((i + 3) - zb, HALF - 1)]; zb += f3;
    __syncthreads();

    // --- stream 1024 rows (16 float4 chunks each) through LDS staging ---
    const float* srcBase = data + (size_t)base * 64;
    for (int outer = 0; outer < OUTERS; ++outer) {
        const int cBase = outer * CHUNKS_PER_OUTER;
#pragma unroll
        for (int k = 0; k < PIPE; ++k) {
            const int c = cBase + k * THREADS + tid;          // coalesced src
            const float* s = srcBase + (size_t)c * 4;
            async_load_b128(s, lds_off_of(&stage[(k * THREADS + tid) * 4]));
        }
        wait_async0();
#pragma unroll
        for (int k = 0; k < PIPE; ++k) {
            const int c    = cBase + k * THREADS + tid;
            const int row  = c >> 4;
            const int part = c & 15;
            float* d = out + (size_t)dests[row] * 64 + part * 4; // 16 lanes -> one 256B row
            async_store_b128(d, lds_off_of(&stage[(k * THREADS + tid) * 4]));
        }
        wait_async0();   // staging reusable next outer
    }
}

// ---------------------------------------------------------------------------
extern "C" void kernel_launch(void* const* d_in, const int* in_sizes, int n_in,
                              void* d_out, int out_size, void* d_ws, size_t ws_size,
                              hipStream_t stream) {
    const float* data  = (const float*)d_in[0];
    const int*   parts = (const int*)d_in[1];
    const int*   idx0  = (const int*)d_in[2];
    const int*   idx1  = (const int*)d_in[3];
    float*       out   = (float*)d_out;

    int* blockSums    = (int*)d_ws;           // 1024 ints
    int* blockOffsets = blockSums + NBLOCKS;  // 1024 ints

    count_kernel  <<<NBLOCKS, THREADS, 0, stream>>>(parts, blockSums);
    scan_kernel   <<<1,       THREADS, 0, stream>>>(blockSums, blockOffsets);
    scatter_kernel<<<NBLOCKS, THREADS, 0, stream>>>(data, parts, idx0, idx1,
                                                    blockOffsets, out);
}